// UpSampling_Padzero_7559142441752
// MI455X (gfx1250) — compile-verified
//
#include <hip/hip_runtime.h>

typedef __attribute__((ext_vector_type(2))) float v2f;
typedef __attribute__((ext_vector_type(8))) float v8f;

#define B_  2
#define C_  128
#define N_  4096
#define M_  16384

// ---------------------------------------------------------------------------
// Kernel 1: pack the WMMA B-operand: for each source point n store
//   {-2x, -2y, -2z, x^2+y^2+z^2}  interleaved as [b][n][4].
// Lane (col, hi) of the argmin wave then does ONE 8-byte load:
//   lo half (hi=0) gets (K0,K1) = (-2x,-2y), hi half gets (K2,K3) = (-2z,nrm)
// matching the 16x16x4 F32 B-matrix VGPR layout (half-wave K split).
// ---------------------------------------------------------------------------
__global__ void pack_b_kernel(const float* __restrict__ xyz,
                              float* __restrict__ bprep) {
    int i = blockIdx.x * blockDim.x + threadIdx.x;   // over B_*N_
    if (i >= B_ * N_) return;
    int b = i / N_, n = i % N_;
    const float* p = xyz + (size_t)b * 3 * N_;
    float x = p[n], y = p[N_ + n], z = p[2 * N_ + n];
    float4 o;
    o.x = -2.0f * x;
    o.y = -2.0f * y;
    o.z = -2.0f * z;
    o.w = x * x + y * y + z * z;
    ((float4*)bprep)[i] = o;
}

// ---------------------------------------------------------------------------
// Kernel 2: one wave per 16-anchor tile. A row m = (ax, ay, az, 1).
// Each v_wmma_f32_16x16x4_f32 produces a 16x16 tile of
//   d'(m,n) = -2 a.x + ||x||^2   (same argmin as full squared distance).
// N-tile loop unrolled x4 with loads hoisted -> 4 global_load_b64 in flight
// per wave (latency hiding at ~2 waves/SIMD occupancy). Running (min,argmin)
// per C-layout slot; branchless xor-shuffle reduction over each 16-lane half.
// ---------------------------------------------------------------------------
__global__ void argmin_wmma_kernel(const float* __restrict__ xyz,
                                   const float* __restrict__ anchor,
                                   const float* __restrict__ bprep,
                                   int* __restrict__ idx_out,
                                   int* __restrict__ exact_out) {
    const int lane = threadIdx.x & 31;
    const int wave = threadIdx.x >> 5;
    const int t    = blockIdx.x * (blockDim.x >> 5) + wave;  // tile id
    const int b    = t >> 10;          // t / (M_/16)
    const int mt   = t & 1023;         // anchor tile within batch
    const int col  = lane & 15;
    const int hi   = lane >> 4;

    // A operand (16x4 f32): lane m = mt*16 + col; VGPR0/1 hold K = 2*hi + 0/1
    const float* ap = anchor + (size_t)b * 3 * M_;
    const int m = mt * 16 + col;
    v2f A;
    A.x = hi ? ap[2 * M_ + m] : ap[m];          // K=2: az   | K=0: ax
    A.y = hi ? 1.0f           : ap[M_ + m];     // K=3: 1.0  | K=1: ay

    float minv[8];
    int   mini[8];
#pragma unroll
    for (int v = 0; v < 8; ++v) { minv[v] = __builtin_inff(); mini[v] = 0; }

    // per-lane B pointer: element (n, 2*hi) of [n][4] interleaved array
    const float* bp = bprep + (size_t)b * N_ * 4 + (size_t)col * 4 + 2 * hi;

    for (int nt = 0; nt < N_ / 16; nt += 4) {
        v2f Bv[4];
#pragma unroll
        for (int u = 0; u < 4; ++u)          // 4 independent loads in flight
            Bv[u] = *(const v2f*)(bp + (size_t)(nt + u) * 16 * 4);
#pragma unroll
        for (int u = 0; u < 4; ++u) {
            v8f c = {};
            // 8 args: (neg_a, A, neg_b, B, c_mod, C, reuse_a, reuse_b)
            v8f d = __builtin_amdgcn_wmma_f32_16x16x4_f32(
                false, A, false, Bv[u], (short)0, c, false, false);
            const int n = (nt + u) * 16 + col;
#pragma unroll
            for (int v = 0; v < 8; ++v) {
                const bool lt = d[v] < minv[v];   // strict < keeps first min
                minv[v] = lt ? d[v] : minv[v];
                mini[v] = lt ? n    : mini[v];
            }
        }
    }

    // branchless min-reduce across the 16 columns of each half-wave
#pragma unroll
    for (int off = 1; off < 16; off <<= 1) {
#pragma unroll
        for (int v = 0; v < 8; ++v) {
            const float od = __shfl_xor(minv[v], off, 32);
            const int   oi = __shfl_xor(mini[v], off, 32);
            const bool take = (od < minv[v]) | ((od == minv[v]) & (oi < mini[v]));
            minv[v] = take ? od : minv[v];
            mini[v] = take ? oi : mini[v];
        }
    }

    if (col == 0) {
        const float* xp = xyz + (size_t)b * 3 * N_;
#pragma unroll
        for (int v = 0; v < 8; ++v) {
            const int mm = mt * 16 + hi * 8 + v;   // D-layout: hi half = rows 8..15
            const int id = mini[v];
            const bool ex = (xp[id]          == ap[mm]) &
                            (xp[N_ + id]     == ap[M_ + mm]) &
                            (xp[2 * N_ + id] == ap[2 * M_ + mm]);
            idx_out[b * M_ + mm]   = id;
            exact_out[b * M_ + mm] = ex ? 1 : 0;
        }
    }
}

// ---------------------------------------------------------------------------
// Kernel 3: gather + zero-fill + anchor passthrough.
// grid (M_/256, C_, B_): threads run over m -> fully coalesced stores;
// feature gathers are random over N but the 4 MB feature array is L2-resident.
// ---------------------------------------------------------------------------
__global__ void gather_kernel(const float* __restrict__ feature,
                              const float* __restrict__ anchor,
                              const int* __restrict__ idx,
                              const int* __restrict__ exact,
                              float* __restrict__ out) {
    const int m = blockIdx.x * blockDim.x + threadIdx.x;
    const int c = blockIdx.y;
    const int b = blockIdx.z;
    const int id = idx[b * M_ + m];
    const int ex = exact[b * M_ + m];
    const float f = feature[(size_t)(b * C_ + c) * N_ + id];
    out[(size_t)B_ * 3 * M_ + (size_t)(b * C_ + c) * M_ + m] = ex ? f : 0.0f;
    if (c == 0) {
#pragma unroll
        for (int k = 0; k < 3; ++k)
            out[(size_t)(b * 3 + k) * M_ + m] = anchor[(size_t)(b * 3 + k) * M_ + m];
    }
}

extern "C" void kernel_launch(void* const* d_in, const int* in_sizes, int n_in,
                              void* d_out, int out_size, void* d_ws, size_t ws_size,
                              hipStream_t stream) {
    const float* xyz     = (const float*)d_in[0];  // [B,3,N]
    const float* feature = (const float*)d_in[1];  // [B,C,N]
    const float* anchor  = (const float*)d_in[2];  // [B,3,M]
    float* out = (float*)d_out;                    // [B,3,M] ++ [B,C,M]

    // workspace layout
    float* bprep = (float*)d_ws;                              // B_*N_*4 floats = 128 KiB
    int*   idx   = (int*)((char*)d_ws + (size_t)B_ * N_ * 4 * sizeof(float));
    int*   exact = idx + B_ * M_;

    // 1) pack WMMA B-operand
    pack_b_kernel<<<(B_ * N_ + 255) / 256, 256, 0, stream>>>(xyz, bprep);

    // 2) WMMA argmin: B_*(M_/16) = 2048 waves, 8 waves (256 thr) per block
    argmin_wmma_kernel<<<(B_ * (M_ / 16)) / 8, 256, 0, stream>>>(
        xyz, anchor, bprep, idx, exact);

    // 3) gather / zero-fill / passthrough
    gather_kernel<<<dim3(M_ / 256, C_, B_), 256, 0, stream>>>(
        feature, anchor, idx, exact, out);
}